// CoreRNN_LN_32143535243750
// MI455X (gfx1250) — compile-verified
//
#include <hip/hip_runtime.h>

// CoreRNN_LN for MI455X (gfx1250): h_t = relu(LN(W_h h_{t-1} + W_g z_t + b_h))
// fp32 WMMA (V_WMMA_F32_16X16X4_F32) for exact fp32 semantics.
// K-tiled GEMM: weights staged into LDS per 16-wide K tile with coalesced
// b128 async-to-LDS loads (ASYNCcnt-tracked), double buffered so the matrix
// pipe never waits on L2 latency. 128 sequential launches = recurrence.

#define T_STEPS 128
#define BATCH   256
#define DG      512
#define DH      1024
#define LN_EPS  1e-5f

#define ROWS    16        // rows per block (WMMA M tile); LN needs whole rows
#define THREADS 512       // 16 waves (wave32)
#define KT      16        // K-tile depth
#define WSTRIDE 20        // 16 + 4 pad floats: 80B rows -> b64-aligned, conflict-free
#define AH_STRIDE 1028    // 1024 + 4 pad
#define AZ_STRIDE 516     // 512 + 4 pad

typedef float v2f __attribute__((ext_vector_type(2)));
typedef float v8f __attribute__((ext_vector_type(8)));
typedef int   v4i __attribute__((ext_vector_type(4)));

typedef __attribute__((address_space(1))) v4i* gptr_b128;   // global (AS1)
typedef __attribute__((address_space(3))) v4i* lptr_b128;   // LDS (AS3)

#if __has_builtin(__builtin_amdgcn_global_load_async_to_lds_b128)
#define USE_ASYNC_LDS 1
#else
#define USE_ASYNC_LDS 0
#endif

static __device__ __forceinline__ void wait_async_zero() {
#if __has_builtin(__builtin_amdgcn_s_wait_asynccnt)
    __builtin_amdgcn_s_wait_asynccnt(0);
#else
    asm volatile("s_wait_asynccnt 0" ::: "memory");
#endif
}

__global__ __launch_bounds__(THREADS)
void rnn_ln_step(const float* __restrict__ h_prev,   // [B, DH] (ignored if first)
                 const float* __restrict__ z_t,      // [B, DG]
                 const float* __restrict__ W_h,      // [DH, DH]
                 const float* __restrict__ W_g,      // [DH, DG]
                 const float* __restrict__ b_h,      // [DH]
                 const float* __restrict__ ln_g,     // [DH]
                 const float* __restrict__ ln_b,     // [DH]
                 float* __restrict__ h_out,          // [B, DH]
                 int first)
{
    // LDS: 2 weight tile buffers [1024][WSTRIDE] + A/z staging + LN stats.
    __shared__ __align__(16) float Wt[2][DH * WSTRIDE];
    __shared__ __align__(16) float Ahz[ROWS * AH_STRIDE + ROWS * AZ_STRIDE];
    __shared__ float mu_s[ROWS], rs_s[ROWS];
    float* Ah = Ahz;                       // h_prev tile [16][1028]
    float* Az = Ahz + ROWS * AH_STRIDE;    // z_t tile    [16][516]

    const int tid      = threadIdx.x;
    const int lane     = tid & 31;
    const int wave     = tid >> 5;         // 0..15
    const int row_base = blockIdx.x * ROWS;

    // ---- stage A operands (h_prev, z_t) into LDS, float4 coalesced ----
    for (int i = tid; i < ROWS * (DG / 4); i += THREADS) {
        int r = i >> 7, c4 = i & 127;
        float4 v = ((const float4*)(z_t + (size_t)(row_base + r) * DG))[c4];
        *(float4*)&Az[r * AZ_STRIDE + c4 * 4] = v;
    }
    if (!first) {
        for (int i = tid; i < ROWS * (DH / 4); i += THREADS) {
            int r = i >> 8, c4 = i & 255;
            float4 v = ((const float4*)(h_prev + (size_t)(row_base + r) * DH))[c4];
            *(float4*)&Ah[r * AH_STRIDE + c4 * 4] = v;
        }
    }

    // ---- WMMA fragment geometry (16x16x4 f32) ----
    const int m      = lane & 15;
    const int khalf  = (lane >> 4) << 1;   // 0 or 2
    const int n      = lane & 15;
    const int n_base = wave * 64;          // this wave's 64-col strip

    v8f acc[4];
    #pragma unroll
    for (int t2 = 0; t2 < 4; ++t2)
        #pragma unroll
        for (int j = 0; j < 8; ++j) acc[t2][j] = 0.0f;

    // Unified K space: tiles [0, whTiles) read W_h/Ah, rest read W_g/Az.
    const int whTiles = first ? 0 : (DH / KT);       // 0 or 64
    const int nTiles  = whTiles + (DG / KT);         // 32 or 96

    // Per-tile weight source row pointer (B col nn = W row nn, K-contiguous).
    auto src_ptr = [&](int t, int nrow, int j) -> const float* {
        if (t < whTiles) return W_h + (size_t)nrow * DH + t * KT + j * 4;
        return W_g + (size_t)nrow * DG + (t - whTiles) * KT + j * 4;
    };

#if !USE_ASYNC_LDS
    float4 pf[8];   // register prefetch staging (fallback path)
#endif

    // fetch: start moving tile t's weights toward LDS buffer `buf`.
    auto fetch = [&](int t, int buf) {
        #pragma unroll
        for (int s = 0; s < 8; ++s) {
            int idx  = tid + s * THREADS;      // 0..4095
            int nrow = idx >> 2;               // 0..1023
            int j    = idx & 3;                // float4 within row segment
            const float* g = src_ptr(t, nrow, j);
#if USE_ASYNC_LDS
            __builtin_amdgcn_global_load_async_to_lds_b128(
                (gptr_b128)g,
                (lptr_b128)&Wt[buf][nrow * WSTRIDE + j * 4],
                0, 0);
#else
            pf[s] = *(const float4*)g;
            (void)buf;
#endif
        }
    };
    // commit: make tile data resident in LDS buffer `buf` (pre-barrier).
    auto commit = [&](int buf) {
#if USE_ASYNC_LDS
        (void)buf;
        wait_async_zero();
#else
        #pragma unroll
        for (int s = 0; s < 8; ++s) {
            int idx  = tid + s * THREADS;
            int nrow = idx >> 2;
            int j    = idx & 3;
            *(float4*)&Wt[buf][nrow * WSTRIDE + j * 4] = pf[s];
        }
#endif
    };

    auto compute_tile = [&](int t, int buf) {
        const float* Ab;
        int As, acol;
        if (t < whTiles) { Ab = Ah; As = AH_STRIDE; acol = t * KT; }
        else             { Ab = Az; As = AZ_STRIDE; acol = (t - whTiles) * KT; }
        const float* wb = &Wt[buf][0];
        #pragma unroll
        for (int kb = 0; kb < 4; ++kb) {
            const int kk = kb * 4 + khalf;
            v2f a;
            a.x = Ab[m * As + acol + kk];
            a.y = Ab[m * As + acol + kk + 1];
            #pragma unroll
            for (int t2 = 0; t2 < 4; ++t2) {
                const int nn = n_base + t2 * 16 + n;
                v2f b;
                b.x = wb[nn * WSTRIDE + kk];
                b.y = wb[nn * WSTRIDE + kk + 1];
                acc[t2] = __builtin_amdgcn_wmma_f32_16x16x4_f32(
                    false, a, false, b, (short)0, acc[t2], false, false);
            }
        }
    };

    // ---- double-buffered K loop ----
    fetch(0, 0);
    commit(0);
    __syncthreads();   // also covers the A staging above
    for (int t = 0; t < nTiles; ++t) {
        const int cur = t & 1;
        if (t + 1 < nTiles) fetch(t + 1, cur ^ 1);
        compute_tile(t, cur);
        if (t + 1 < nTiles) commit(cur ^ 1);
        __syncthreads();   // buf[cur] free for reuse; buf[cur^1] visible to all
    }

    // ---- spill pre-activation (+bias) to LDS (alias dead Ah region) ----
    float* pre = Ahz;   // [ROWS][AH_STRIDE]
    #pragma unroll
    for (int t2 = 0; t2 < 4; ++t2) {
        const int nn   = n_base + t2 * 16 + n;
        const float bv = b_h[nn];
        #pragma unroll
        for (int r = 0; r < 8; ++r) {
            const int mm = (lane < 16) ? r : (r + 8);
            pre[mm * AH_STRIDE + nn] = acc[t2][r] + bv;
        }
    }
    __syncthreads();

    // ---- LayerNorm stats: wave w reduces row w ----
    {
        float s = 0.0f, ss = 0.0f;
        for (int c = lane; c < DH; c += 32) {
            float x = pre[wave * AH_STRIDE + c];
            s += x; ss += x * x;
        }
        #pragma unroll
        for (int o = 16; o > 0; o >>= 1) {
            s  += __shfl_xor(s,  o, 32);
            ss += __shfl_xor(ss, o, 32);
        }
        if (lane == 0) {
            float mu  = s  * (1.0f / DH);
            float var = ss * (1.0f / DH) - mu * mu;
            mu_s[wave] = mu;
            rs_s[wave] = rsqrtf(var + LN_EPS);
        }
    }
    __syncthreads();

    // ---- normalize + relu + coalesced store ----
    for (int i = tid; i < ROWS * DH; i += THREADS) {
        const int r = i >> 10;
        const int c = i & (DH - 1);
        float x = (pre[r * AH_STRIDE + c] - mu_s[r]) * rs_s[r] * ln_g[c] + ln_b[c];
        h_out[(size_t)(row_base + r) * DH + c] = fmaxf(x, 0.0f);
    }
}

extern "C" void kernel_launch(void* const* d_in, const int* in_sizes, int n_in,
                              void* d_out, int out_size, void* d_ws, size_t ws_size,
                              hipStream_t stream) {
    const float* z_seq = (const float*)d_in[0];   // [T,B,DG]
    const float* W_h   = (const float*)d_in[1];   // [DH,DH]
    const float* W_g   = (const float*)d_in[2];   // [DH,DG]
    const float* b_h   = (const float*)d_in[3];   // [DH]
    const float* ln_g  = (const float*)d_in[4];   // [DH]
    const float* ln_b  = (const float*)d_in[5];   // [DH]
    float* out = (float*)d_out;                   // [T,B,DH]

    (void)d_ws; (void)ws_size; (void)in_sizes; (void)n_in; (void)out_size;

    for (int t = 0; t < T_STEPS; ++t) {
        const float* h_prev = (t == 0) ? out : (out + (size_t)(t - 1) * BATCH * DH);
        rnn_ln_step<<<BATCH / ROWS, THREADS, 0, stream>>>(
            h_prev,
            z_seq + (size_t)t * BATCH * DG,
            W_h, W_g, b_h, ln_g, ln_b,
            out + (size_t)t * BATCH * DH,
            (t == 0) ? 1 : 0);
    }
}